// FixedTrajectoryGenerator_36515811950787
// MI455X (gfx1250) — compile-verified
//
#include <hip/hip_runtime.h>
#include <hip/hip_bf16.h>

// Live computation of the reference (everything else is dead code):
//   out[b, t, a] = start_state[b, a] + (t/255) * patterns[instruction_id[b], t, a]
// B=256, T=256, AD=64, V=1000. Pure streaming FMA: ~32 MiB of traffic,
// ~1.4 us at 23.3 TB/s HBM. No matrix ops survive dead-code elimination, so
// WMMA is not applicable; we optimize the CDNA5 data-movement path instead:
//  - 128-bit coalesced global loads/stores
//  - non-temporal th hints on the one-touch streams (patterns, output)
//  - async global->LDS (ASYNCcnt) staging of the reused start_state row

typedef float __attribute__((ext_vector_type(4))) v4f;

#define TRAJ_T      256
#define TRAJ_B      256
#define TRAJ_AD     64
#define F4_PER_ROW  16   // AD / 4
#define TROWS_PER_BLK 16 // t-rows handled per 256-thread block

__global__ __launch_bounds__(256) void FixedTrajectoryGenerator_kernel(
    const v4f* __restrict__ start4,   // (B, 16) float4
    const int* __restrict__ instr,    // (B,)
    const v4f* __restrict__ pat4,     // (V, T, 16) float4
    v4f*       __restrict__ out4)     // (B, T, 16) float4
{
    // 8 waves * 16 float4 * 16 B = 2 KiB: one private start-row copy per wave
    __shared__ __align__(16) char smem[8 * F4_PER_ROW * 16];

    const int blk    = blockIdx.x;          // 0 .. 4095
    const int tid    = threadIdx.x;         // 0 .. 255
    const int b      = blk >> 4;            // batch row (uniform per block)
    const int tchunk = blk & 15;            // which 16-t slice
    const int trow   = tid >> 4;            // 0 .. 15
    const int a4     = tid & 15;            // float4 column within action dim
    const int t      = (tchunk << 4) | trow;

    const int wave = tid >> 5;
    const int lane = tid & 31;

    const uint32_t lds0  = (uint32_t)(uintptr_t)(&smem[0]);
    const uint32_t wbase = lds0 + ((uint32_t)wave << 8); // 256 B per wave

    // --- CDNA5 async global->LDS stage of start_state[b] (64 floats) -------
    // Lanes 0..15 of each wave each move one float4; tracked by ASYNCcnt.
    if (lane < 16) {
        const uint64_t ga = (uint64_t)(uintptr_t)(start4 + ((size_t)b << 4) + lane);
        const uint32_t la = wbase + ((uint32_t)lane << 4);
        asm volatile("global_load_async_to_lds_b128 %0, %1, off"
                     :: "v"(la), "v"(ga)
                     : "memory");
    }
    // Wave-private copy => only the issuing wave's ASYNCcnt matters; no barrier.
    asm volatile("s_wait_asynccnt 0x0" ::: "memory");

    // LDS readback; fold the dependent dscnt wait into the same asm so no use
    // can be scheduled between the ds_load and its wait.
    v4f s;
    {
        const uint32_t ra = wbase + ((uint32_t)a4 << 4);
        asm volatile("ds_load_b128 %0, %1\n\t"
                     "s_wait_dscnt 0x0"
                     : "=v"(s) : "v"(ra) : "memory");
    }

    // Block-uniform gather index -> scalar load.
    const int   id = instr[b];
    const float w  = (float)t * (1.0f / 255.0f);   // linspace(0,1,256)[t]

    // One-touch stream: non-temporal load (keep WGP$/L2 clean; no reuse).
    const v4f p = __builtin_nontemporal_load(
        pat4 + (((size_t)id) << 12) + ((size_t)tchunk << 8) + (size_t)tid);

    const v4f r = s + w * p;

    // Output is write-once, never re-read: non-temporal store.
    __builtin_nontemporal_store(r, out4 + (((size_t)blk) << 8) + (size_t)tid);
}

extern "C" void kernel_launch(void* const* d_in, const int* in_sizes, int n_in,
                              void* d_out, int out_size, void* d_ws, size_t ws_size,
                              hipStream_t stream) {
    (void)in_sizes; (void)n_in; (void)out_size; (void)d_ws; (void)ws_size;

    const v4f* start4 = (const v4f*)d_in[0];   // start_state (B, SD=64) f32
    const int* instr  = (const int*)d_in[1];   // instruction_id (B,) i32
    const v4f* pat4   = (const v4f*)d_in[27];  // patterns (V, T, AD) f32
    v4f*       out4   = (v4f*)d_out;           // (B, T, AD) f32

    const int blocks = TRAJ_B * (TRAJ_T / TROWS_PER_BLK); // 4096
    FixedTrajectoryGenerator_kernel<<<dim3(blocks), dim3(256), 0, stream>>>(
        start4, instr, pat4, out4);
}